// MeanAggregator_22299470201189
// MI455X (gfx1250) — compile-verified
//
#include <hip/hip_runtime.h>
#include <stdint.h>

// MeanAggregator (GraphSAGE-style): out[n] = mean over UNIQUE ids of
// {to_neighs[n,0..31], nodes[n]} of features[id].  N=50000, K=32, D=128.
// Bandwidth-bound gather (~880 MB -> ~38us floor @ 23.3 TB/s).
// CDNA5 path: GLOBAL_LOAD_ASYNC_TO_LDS_B128 double(4x)-buffered row pipeline.

#define KN    32      // sampled neighbors per node
#define KP1   33      // +self
#define DF    128     // feature dim
#define WPB   8       // waves per block
#define NBUF  4       // async pipeline depth (rows in flight per wave)

__device__ __forceinline__ void async_row_load(const float* g, unsigned ldsoff) {
  // s_wait_dscnt 0 guards the (tiny) window where a still-pending ds_read of
  // the recycled buffer could observe the async engine's overwrite.
  asm volatile("s_wait_dscnt 0x0\n\t"
               "global_load_async_to_lds_b128 %0, %1, off"
               :: "v"(ldsoff), "v"(g)
               : "memory");
}

__device__ __forceinline__ void wait_async_le(int n) {
  if (n <= 0)      asm volatile("s_wait_asynccnt 0x0" ::: "memory");
  else if (n == 1) asm volatile("s_wait_asynccnt 0x1" ::: "memory");
  else if (n == 2) asm volatile("s_wait_asynccnt 0x2" ::: "memory");
  else             asm volatile("s_wait_asynccnt 0x3" ::: "memory");
}

__global__ __launch_bounds__(256) void mean_agg_kernel(
    const int* __restrict__ nodes,
    const int* __restrict__ to_neighs,
    const float* __restrict__ features,
    float* __restrict__ out,
    int n_nodes)
{
  __shared__ int    s_ids[WPB][40];          // 33 ids per wave (+pad)
  __shared__ float4 s_buf[WPB][NBUF][32];    // 4 x 512B row buffers per wave

  const int w    = threadIdx.x >> 5;
  const int l    = threadIdx.x & 31;
  const int node = blockIdx.x * WPB + w;
  const bool active = node < n_nodes;

  if (active) {
    s_ids[w][l] = to_neighs[(size_t)node * KN + l];
    if (l == 0) s_ids[w][KN] = nodes[node];
  }
  __syncthreads();
  if (!active) return;

  // --- duplicate detection: id j (j=lane+1 in 1..32) is dup iff exists i<j equal
  int myid = s_ids[w][l + 1];
  bool dup = false;
  #pragma unroll
  for (int i = 0; i < 32; ++i)
    dup = dup || ((i <= l) && (s_ids[w][i] == myid));   // i <= l  <=>  i < j
  unsigned dmask = (unsigned)__ballot(dup ? 1 : 0);     // bit (j-1) set => row j dup
  int cnt = KP1 - __popc(dmask);                        // unique count

  // LDS byte offset of this lane's 16B slot in buffer 0 (flat addr low 32 bits
  // == group-segment-relative LDS address per ISA 10.2).
  unsigned loff = (unsigned)(uintptr_t)(&s_buf[w][0][0]) + (unsigned)(l * 16);

  // --- prologue: fill the pipeline with rows 0..NBUF-1
  #pragma unroll
  for (int p = 0; p < NBUF; ++p) {
    int id = s_ids[w][p];
    const float* g = features + ((size_t)(unsigned)id << 7) + (l << 2);
    async_row_load(g, loff + p * 512);
  }

  float4 acc = make_float4(0.f, 0.f, 0.f, 0.f);
  #pragma unroll
  for (int j = 0; j < KP1; ++j) {
    // rows complete in order: allow min(32-j, NBUF-1) still outstanding
    int wn = (KP1 - 1 - j) < (NBUF - 1) ? (KP1 - 1 - j) : (NBUF - 1);
    wait_async_le(wn);
    float4 v = s_buf[w][j & (NBUF - 1)][l];             // ds_load_b128
    float wj = (j > 0 && ((dmask >> (j - 1)) & 1u)) ? 0.f : 1.f;
    acc.x += wj * v.x;
    acc.y += wj * v.y;
    acc.z += wj * v.z;
    acc.w += wj * v.w;
    if (j + NBUF < KP1) {                               // refill recycled buffer
      int id = s_ids[w][j + NBUF];
      const float* g = features + ((size_t)(unsigned)id << 7) + (l << 2);
      async_row_load(g, loff + ((j + NBUF) & (NBUF - 1)) * 512);
    }
  }

  float s = 1.0f / (float)cnt;
  float4 r = make_float4(acc.x * s, acc.y * s, acc.z * s, acc.w * s);
  *((float4*)(out + (size_t)node * DF) + l) = r;        // coalesced 512B/wave store
}

extern "C" void kernel_launch(void* const* d_in, const int* in_sizes, int n_in,
                              void* d_out, int out_size, void* d_ws, size_t ws_size,
                              hipStream_t stream) {
  const int*   nodes     = (const int*)d_in[0];
  const int*   to_neighs = (const int*)d_in[1];
  const float* features  = (const float*)d_in[2];
  float*       out       = (float*)d_out;
  const int n = in_sizes[0];
  const int blocks = (n + WPB - 1) / WPB;
  mean_agg_kernel<<<blocks, WPB * 32, 0, stream>>>(nodes, to_neighs, features, out, n);
}